// WeightedRankingLoss_67654324846806
// MI455X (gfx1250) — compile-verified
//
#include <hip/hip_runtime.h>
#include <stdint.h>

#define N_PTS   8192
#define D_DIM   128
#define N_ANCH  1024
#define K_SEL   64
#define MARGIN_F 1.0f
#define THREADS 256
#define WAVES   (THREADS / 32)

typedef __attribute__((ext_vector_type(2))) float v2f;
typedef __attribute__((ext_vector_type(8))) float v8f;

// One workgroup per (side, anchor-row).
//  side 0: a1 = out1[anchor1[row]] vs all of out2, weights ot[anchor1[row], j]
//  side 1: a2 = out2[anchor2[row]] vs all of out1, weights ot[j, anchor2[row]]
__global__ __launch_bounds__(THREADS)
void wrl_row_kernel(const float* __restrict__ out1,
                    const float* __restrict__ out2,
                    const float* __restrict__ ot_cost,
                    const int*   __restrict__ anchor1,
                    const int*   __restrict__ anchor2,
                    float*       __restrict__ partials)
{
    __shared__ float        sA[D_DIM];
    __shared__ float        sB[D_DIM];
    __shared__ unsigned int keys[N_PTS];      // orderable f32 bits per candidate
    __shared__ unsigned long long redk[WAVES];
    __shared__ int          negIdx[K_SEL];
    __shared__ float        waveAcc[WAVES];
    __shared__ float        sPos;

    const int tid  = threadIdx.x;
    const int lane = tid & 31;
    const int wave = tid >> 5;

    const int row  = blockIdx.x & (N_ANCH - 1);
    const int side = blockIdx.x >> 10;

    const int aSel = (side == 0) ? anchor1[row] : anchor2[row];
    const int bSel = (side == 0) ? anchor2[row] : anchor1[row];
    const float* aBase = ((side == 0) ? out1 : out2) + (size_t)aSel * D_DIM;
    const float* bBase = ((side == 0) ? out2 : out1) + (size_t)bSel * D_DIM;
    const float* cand  = (side == 0) ? out2 : out1;

    if (tid < D_DIM) {
        sA[tid] = aBase[tid];
        sB[tid] = bBase[tid];
    }
    __syncthreads();

    // ---- pos = sum_d |a1 - a2| (same value for both sides) ----
    if (wave == 0) {
        float p = 0.f;
        for (int d = lane; d < D_DIM; d += 32) p += fabsf(sA[d] - sB[d]);
        for (int off = 16; off; off >>= 1) p += __shfl_xor(p, off, 32);
        if (lane == 0) sPos = p;
    }

    // ---- L1 distance row: V_WMMA_F32_16X16X4_F32, B = ones, C accumulates ----
    // A (16x4 f32) layout: lanes 0-15 hold K={0,1} for M=lane, lanes 16-31 hold
    // K={2,3} for M=lane-16.  D[m,n] = sum_k |a[k]-cand[m,k]| for every n.
    // Two independent accumulator chains (even/odd K-chunks) to halve the
    // WMMA->WMMA RAW dependency depth.
    const int m    = lane & 15;
    const int kgrp = lane >> 4;             // 0 or 1
    const v2f bOnes = {1.0f, 1.0f};

    for (int tile = wave; tile < N_PTS / 16; tile += WAVES) {
        const int j0 = tile * 16;
        const float* crow = cand + (size_t)(j0 + m) * D_DIM + 2 * kgrp;
        v8f c0 = {0.f, 0.f, 0.f, 0.f, 0.f, 0.f, 0.f, 0.f};
        v8f c1 = {0.f, 0.f, 0.f, 0.f, 0.f, 0.f, 0.f, 0.f};
        #pragma unroll
        for (int d0 = 0; d0 < D_DIM; d0 += 8) {
            // chunk A: dims [d0, d0+4)
            {
                const float y0 = crow[d0 + 0];
                const float y1 = crow[d0 + 1];
                const float x0 = sA[d0 + 2 * kgrp + 0];
                const float x1 = sA[d0 + 2 * kgrp + 1];
                v2f a;
                a.x = fabsf(x0 - y0);
                a.y = fabsf(x1 - y1);
                c0 = __builtin_amdgcn_wmma_f32_16x16x4_f32(
                        false, a, false, bOnes, (short)0, c0, false, false);
            }
            // chunk B: dims [d0+4, d0+8)
            {
                const float y0 = crow[d0 + 4];
                const float y1 = crow[d0 + 5];
                const float x0 = sA[d0 + 4 + 2 * kgrp + 0];
                const float x1 = sA[d0 + 4 + 2 * kgrp + 1];
                v2f a;
                a.x = fabsf(x0 - y0);
                a.y = fabsf(x1 - y1);
                c1 = __builtin_amdgcn_wmma_f32_16x16x4_f32(
                        false, a, false, bOnes, (short)0, c1, false, false);
            }
        }
        // Lane 0 holds D for M=0..7 in c[0..7]; lane 16 holds M=8..15.
        if ((lane & 15) == 0) {
            const int base = j0 + (kgrp << 3);
            #pragma unroll
            for (int q = 0; q < 8; ++q)
                keys[base + q] = __float_as_uint(c0[q] + c1[q]); // dist>=0 -> orderable
        }
    }
    __syncthreads();

    // ---- top-K: 64 arg-min passes (min dist-bits, tie -> min index) ----
    for (int t = 0; t < K_SEL; ++t) {
        unsigned long long best = ~0ull;
        for (int idx = tid; idx < N_PTS; idx += THREADS) {
            unsigned long long k =
                ((unsigned long long)keys[idx] << 32) | (unsigned int)idx;
            best = (k < best) ? k : best;
        }
        for (int off = 16; off; off >>= 1) {
            unsigned long long o = __shfl_xor(best, off, 32);
            best = (o < best) ? o : best;
        }
        if (lane == 0) redk[wave] = best;
        __syncthreads();
        if (tid == 0) {
            unsigned long long w = redk[0];
            for (int q = 1; q < WAVES; ++q)
                w = (redk[q] < w) ? redk[q] : w;
            const int j = (int)(w & 0xffffffffu);
            negIdx[t] = j;
            keys[j] = 0xffffffffu;   // exclude from later passes
        }
        __syncthreads();
    }

    // ---- weighted-neighbor loss:  relu(pos + margin - sum_d |a - w*neg|) ----
    const float pos = sPos;
    float acc = 0.f;
    for (int k = wave; k < K_SEL; k += WAVES) {
        const int j = negIdx[k];
        const float w = (side == 0)
                        ? ot_cost[(size_t)aSel * N_PTS + j]
                        : ot_cost[(size_t)j * N_PTS + aSel];
        const float* nrow = cand + (size_t)j * D_DIM;
        float s = 0.f;
        for (int d = lane; d < D_DIM; d += 32)
            s += fabsf(sA[d] - w * nrow[d]);
        for (int off = 16; off; off >>= 1) s += __shfl_xor(s, off, 32);
        if (lane == 0) acc += fmaxf(pos + MARGIN_F - s, 0.f);
    }
    if (lane == 0) waveAcc[wave] = acc;
    __syncthreads();
    if (tid == 0) {
        float tot = 0.f;
        for (int q = 0; q < WAVES; ++q) tot += waveAcc[q];
        partials[blockIdx.x] = tot;
    }
}

// Deterministic fixed-order final reduction of 2048 partials -> scalar.
__global__ __launch_bounds__(THREADS)
void wrl_finish(const float* __restrict__ partials, float* __restrict__ out)
{
    __shared__ float red[THREADS];
    const int tid = threadIdx.x;
    float s = 0.f;
    for (int i = tid; i < 2 * N_ANCH; i += THREADS) s += partials[i];
    red[tid] = s;
    __syncthreads();
    for (int off = THREADS / 2; off; off >>= 1) {
        if (tid < off) red[tid] += red[tid + off];
        __syncthreads();
    }
    if (tid == 0) out[0] = red[0] / (float)(N_ANCH * K_SEL);
}

extern "C" void kernel_launch(void* const* d_in, const int* in_sizes, int n_in,
                              void* d_out, int out_size, void* d_ws, size_t ws_size,
                              hipStream_t stream)
{
    const float* out1 = (const float*)d_in[0];
    const float* out2 = (const float*)d_in[1];
    const float* ot   = (const float*)d_in[2];
    const int*   an1  = (const int*)d_in[3];
    const int*   an2  = (const int*)d_in[4];
    float* partials   = (float*)d_ws;          // 2048 floats = 8 KB scratch

    wrl_row_kernel<<<2 * N_ANCH, THREADS, 0, stream>>>(out1, out2, ot, an1, an2,
                                                       partials);
    wrl_finish<<<1, THREADS, 0, stream>>>(partials, (float*)d_out);
}